// WaveletRefinement_7825430413854
// MI455X (gfx1250) — compile-verified
//
#include <hip/hip_runtime.h>
#include <math.h>

typedef float v2f __attribute__((ext_vector_type(2)));
typedef float v8f __attribute__((ext_vector_type(8)));
typedef float f4  __attribute__((ext_vector_type(4)));

#define INV_SQRT2F 0.70710678118654752440f

// ---------------------------------------------------------------------------
// Kernel G: 16xK @ (NxK)^T + bias, activation. One wave per 16x16 output tile.
// V_WMMA_F32_16X16X4_F32 (CDNA5).  Lane layout per ISA 7.12.2:
//   A: lanes 0-15 M=lane {K=0,1}, lanes 16-31 M=lane-16 {K=2,3}
//   B: lanes 0-15 N=lane {K=0,1}, lanes 16-31 N=lane-16 {K=2,3}
//   D: vgpr i -> M = i + 8*(lane>=16), N = lane&15
// ---------------------------------------------------------------------------
__global__ __launch_bounds__(32)
void gemm16_wmma(const float* __restrict__ X, const float* __restrict__ W,
                 const float* __restrict__ bias, float* __restrict__ out,
                 int act /*0=silu,1=sigmoid*/)
{
    const int K = 1024;
    int lane = threadIdx.x;
    int m    = lane & 15;
    int half = lane >> 4;
    int j0   = blockIdx.x << 4;

    v8f c = {0.f,0.f,0.f,0.f,0.f,0.f,0.f,0.f};
    const float* xrow = X + m * K + 2 * half;
    const float* wrow = W + (j0 + m) * K + 2 * half;   // B[kk][n] = W[j0+n][k+kk]
    for (int k = 0; k < K; k += 4) {
        v2f a = *(const v2f*)(xrow + k);
        v2f b = *(const v2f*)(wrow + k);
        c = __builtin_amdgcn_wmma_f32_16x16x4_f32(false, a, false, b,
                                                  (short)0, c, false, false);
    }
    int   col  = j0 + m;
    float bcol = bias[col];
#pragma unroll
    for (int i = 0; i < 8; ++i) {
        int   row = i + 8 * half;
        float x   = c[i] + bcol;
        float s   = 1.0f / (1.0f + __expf(-x));
        float r   = act ? s : x * s;
        out[row * 1024 + col] = r;
    }
}

// ---------------------------------------------------------------------------
// Kernel S: lo_6[b][k6][d] = (1/8) * sum_{n in (64(k6-1), 64*k6]} v[b][n][d]
// Each lane owns 4 contiguous d columns -> all traffic is global b128.
// Block covers d = 0..1023 for one (b, k6).
// ---------------------------------------------------------------------------
__global__ __launch_bounds__(256)
void haar_block_sum(const float* __restrict__ v, float* __restrict__ lo6)
{
    int bid = blockIdx.x;           // 16 * 64 blocks
    int k6  = bid & 63;
    int b   = bid >> 6;
    int d   = threadIdx.x << 2;     // 256 lanes * 4 floats = 1024

    int nstart = (k6 << 6) - 63;    // window = [64*k6-63, 64*k6], circular
    const float* base = v + ((size_t)b << 12) * 1024 + d;
    f4 acc = {0.f,0.f,0.f,0.f};
    for (int i = 0; i < 64; ++i) {
        int n = (nstart + i) & 4095;
        acc += *(const f4*)(base + (size_t)n * 1024);
    }
    *(f4*)(lo6 + (((size_t)b << 6) + k6) * 1024 + d) = acc * 0.125f; // 2^{-3}
}

// ---------------------------------------------------------------------------
// Kernel C: coarse levels per column. Build lo_7..lo_11 from lo_6, then the
// e-recursion down from e_10 (e_11 == 0 analytically) to e_6, in place.
// Tiny kernel (8 MB total traffic, L2-resident) - scalar is fine.
// ---------------------------------------------------------------------------
__global__ __launch_bounds__(256)
void haar_coarse(const float* __restrict__ lo6g, float* __restrict__ e6g)
{
    int tid = blockIdx.x * 256 + threadIdx.x;   // 16384 columns
    int b = tid >> 10, d = tid & 1023;

    float buf[126];  // lo6[0..63] lo7[64..95] lo8[96..111] lo9[112..119]
                     // lo10[120..123] lo11[124..125]
    const int offs[6] = {0, 64, 96, 112, 120, 124};

    const float* src = lo6g + ((size_t)b << 6) * 1024 + d;
    for (int k = 0; k < 64; ++k) buf[k] = src[(size_t)k * 1024];

    for (int lev = 0; lev < 5; ++lev) {         // build parents (levels 7..11)
        int offC = offs[lev], offP = offs[lev + 1];
        int L = 64 >> lev;
        for (int k = 0; k < (L >> 1); ++k) {
            float a  = buf[offC + ((2 * k - 1) & (L - 1))];
            float bb = buf[offC + 2 * k];
            buf[offP + k] = (a + bb) * INV_SQRT2F;
        }
    }
    for (int lev = 4; lev >= 0; --lev) {        // e_10 (lev=4) ... e_6 (lev=0)
        int off = offs[lev], offP = offs[lev + 1];
        int L = 64 >> lev;
        float wrapA = buf[off + L - 1];         // lo[L-1], consumed by k==1
        for (int k = L - 1; k >= 0; --k) {      // descending: in-place safe
            float parent = (lev == 4) ? 0.f : buf[offP + (k >> 1)];
            float val = parent * INV_SQRT2F;
            if (k & 1) {
                float lo_km2 = (k >= 2) ? buf[off + k - 2] : wrapA;
                val += lo_km2 - buf[off + k];
            }
            buf[off + k] = val;
        }
    }
    float* dst = e6g + ((size_t)b << 6) * 1024 + d;
    for (int k = 0; k < 64; ++k) dst[(size_t)k * 1024] = buf[k];
}

// ---------------------------------------------------------------------------
// Kernel M: main pass.  Block = (b, 256-n tile, 32-d chunk).
// All LDS/global traffic as 128-bit ops: LDS prefix array stored as float4
// rows (stride 8 x float4 = 32 floats).  Prefix sums per column give every
// dyadic window sum in 2 reads and return x[n] itself, so v is read once
// (the 96-row halo overlaps neighbors' main ranges -> L2 hits).
// Output stores are non-temporal: streamed once, keep L2 for halo sharing.
// ---------------------------------------------------------------------------
__global__ __launch_bounds__(256)
void haar_apply(const float* __restrict__ v, const float* __restrict__ e6g,
                const float* __restrict__ gate, const int* __restrict__ on_mask,
                float* __restrict__ out)
{
    __shared__ f4 P4[353 * 8];     // [i][cg]: prefix over x, 4 cols per slot
    __shared__ f4 seg4[8 * 8];

    int bid    = blockIdx.x;       // 16 * 16 * 32 blocks
    int dchunk = bid & 31;
    int nblk   = (bid >> 5) & 15;
    int b      = bid >> 9;
    int n0     = nblk << 8;
    int d0     = dchunk << 5;
    int nb     = n0 - 96;          // halo: 3*2^5 for r<=5 windows
    int tid    = threadIdx.x;

    const float* vb = v + ((size_t)b << 12) * 1024;
    for (int idx = tid; idx < 352 * 8; idx += 256) {  // x -> P4 rows 1..352
        int i = idx >> 3, cg = idx & 7;
        int n = (nb + i) & 4095;                      // circular
        P4[(i + 1) * 8 + cg] = *(const f4*)(vb + (size_t)n * 1024 + d0 + cg * 4);
    }
    if (tid < 8) P4[tid] = (f4){0.f,0.f,0.f,0.f};
    __syncthreads();

    // per-column prefix scan: 64 threads = 8 segs x 8 col-groups, 44 rows each
    if (tid < 64) {
        int cg = tid & 7, seg = tid >> 3, base = 1 + seg * 44;
        f4 s = {0.f,0.f,0.f,0.f};
        for (int ii = 0; ii < 44; ++ii) s += P4[(base + ii) * 8 + cg];
        seg4[seg * 8 + cg] = s;
    }
    __syncthreads();
    if (tid < 8) {
        f4 off = {0.f,0.f,0.f,0.f};
        for (int seg = 0; seg < 8; ++seg) {
            f4 t = seg4[seg * 8 + tid];
            seg4[seg * 8 + tid] = off;
            off += t;
        }
    }
    __syncthreads();
    if (tid < 64) {
        int cg = tid & 7, seg = tid >> 3, base = 1 + seg * 44;
        f4 s = seg4[seg * 8 + cg];
        for (int ii = 0; ii < 44; ++ii) {
            s += P4[(base + ii) * 8 + cg];
            P4[(base + ii) * 8 + cg] = s;
        }
    }
    __syncthreads();

    float msk = on_mask[b] ? 1.f : 0.f;
    const float C5 = 0.17677669529663688f;  // (1/sqrt2)^5 for e_6 seed

    // tasks: 128 k1 x 8 col-groups = 1024, 4 iterations of 256 threads
    for (int task = tid; task < 128 * 8; task += 256) {
        int cg = task & 7, kl = task >> 3;
        int d  = d0 + cg * 4;
        int k1 = (n0 >> 1) + kl;

        f4 acc = *(const f4*)(e6g + (((size_t)b << 6) + (k1 >> 5)) * 1024 + d);
        acc *= C5;
#pragma unroll
        for (int r = 1; r <= 5; ++r) {
            int kr = k1 >> (r - 1);
            if (kr & 1) {
                int mB = kr << r;
                int iB = mB - nb + 1;
                int iA = iB - (2 << r);     // shift window by 2*2^r
                int w  = 1 << r;
                f4 SA = P4[iA * 8 + cg] - P4[(iA - w) * 8 + cg];
                f4 SB = P4[iB * 8 + cg] - P4[(iB - w) * 8 + cg];
                // (1/sqrt2)^{r-1} * 2^{-r/2} = sqrt(2)/2^r
                acc += (1.41421356237309515f / (float)(1 << r)) * (SA - SB);
            }
        }
        f4 e1h = acc * INV_SQRT2F;                    // e_1[k1]/sqrt2
        f4 g   = *(const f4*)(gate + (size_t)b * 1024 + d) * msk;

        int me = 2 * k1;            // even output n (< 4096, no wrap)
        int ie = me - nb;           // local coordinate in [96, 352)
        f4 Pm  = P4[(ie - 1) * 8 + cg];
        f4 P0  = P4[ ie      * 8 + cg];
        f4 P1  = P4[(ie + 1) * 8 + cg];
        f4 P2  = P4[(ie + 2) * 8 + cg];
        f4 xe  = P1 - P0;           // x[2k1]
        f4 xo  = P2 - P1;           // x[2k1+1]
        f4 xom = P0 - Pm;           // x[2k1-1]

        f4 ye = xe + (xe  + e1h) * g;   // out = v + vref*gate*mask
        f4 yo = xo + (xom + e1h) * g;

        float* op = out + (((size_t)b << 12) + me) * 1024 + d;
        __builtin_nontemporal_store(ye, (f4*)op);
        __builtin_nontemporal_store(yo, (f4*)(op + 1024));
    }
}

// ---------------------------------------------------------------------------
extern "C" void kernel_launch(void* const* d_in, const int* in_sizes, int n_in,
                              void* d_out, int out_size, void* d_ws, size_t ws_size,
                              hipStream_t stream)
{
    const float* v    = (const float*)d_in[0];   // [16,4096,1024]
    const float* q    = (const float*)d_in[1];   // [16,1024]
    const int*   mask = (const int*)  d_in[2];   // [16]
    const float* W1   = (const float*)d_in[3];   // [1024,1024]
    const float* b1   = (const float*)d_in[4];   // [1024]
    const float* W2   = (const float*)d_in[5];   // [1024,1024]
    const float* b2   = (const float*)d_in[6];   // [1024]
    float* out = (float*)d_out;

    // workspace: h(16K) | gate(16K) | lo6(1M) | e6(1M) floats  (~8.5 MB)
    float* ws   = (float*)d_ws;
    float* h    = ws;
    float* gate = ws + 16384;
    float* lo6  = ws + 32768;
    float* e6   = lo6 + (size_t)16 * 64 * 1024;

    gemm16_wmma   <<<64,           32, 0, stream>>>(q, W1, b1, h,    0); // silu
    gemm16_wmma   <<<64,           32, 0, stream>>>(h, W2, b2, gate, 1); // sigmoid
    haar_block_sum<<<16 * 64,     256, 0, stream>>>(v, lo6);
    haar_coarse   <<<64,          256, 0, stream>>>(lo6, e6);
    haar_apply    <<<16 * 16 * 32,256, 0, stream>>>(v, e6, gate, mask, out);
}